// RandomPixelMapping_19593640805006
// MI455X (gfx1250) — compile-verified
//
#include <hip/hip_runtime.h>
#include <stdint.h>

// out[b,c,h,w] = table[b,c, clamp(rne(x*255), 0, 255)]
// Pure bandwidth kernel: ~403 MB HBM traffic -> ~17us at 23.3 TB/s.
// CDNA5 paths used: TDM tensor_load_to_lds for table staging (ASYNC DMA,
// s_wait_tensorcnt), LDS gather (ds_load_b32), b128 nontemporal streaming IO.

typedef __attribute__((ext_vector_type(4))) float    v4f;
typedef __attribute__((ext_vector_type(4))) uint32_t u32x4;
typedef __attribute__((ext_vector_type(8))) int      i32x8;
typedef __attribute__((ext_vector_type(4))) int      i32x4;

#define LUT_N 256
#define IMG_HW (512 * 512)
#define N_SLICES (64 * 3)
#define TPB 256
#define V4_PER_THREAD 4
#define ELEMS_PER_BLOCK (TPB * 4 * V4_PER_THREAD)     /* 4096 */
#define CHUNKS_PER_SLICE (IMG_HW / ELEMS_PER_BLOCK)    /* 64   */

__global__ __launch_bounds__(TPB)
void RandomPixelMapping_lut_kernel(const float* __restrict__ x,
                                   const float* __restrict__ table,
                                   float* __restrict__ out) {
    __shared__ float lut[LUT_N];

    const uint32_t slice = blockIdx.y;   // b*3 + c, 0..191
    const uint32_t chunk = blockIdx.x;   // 0..63

    // --- Stage this (b,c) table slice (1 KB) into LDS via the Tensor Data
    // Mover. Tensor ops ignore EXEC and issue once per wave, so only wave 0
    // of the block enters this region; the workgroup barrier below publishes
    // the LDS data to the other 7 waves.
    if (threadIdx.x < 32) {
        const uint64_t gaddr  = (uint64_t)(uintptr_t)(table + (size_t)slice * LUT_N);
        // Flat LDS addresses are {SHARED_BASE, offset[31:0]} -> low 32 bits
        // are the workgroup-relative LDS byte offset the D# wants.
        const uint32_t ldsoff = (uint32_t)(uintptr_t)(void*)&lut[0];

        // D# group 0: count=1 (valid user descriptor), lds_addr,
        // 57-bit global_addr in bits [120:64], type=2 ("image") in [127:126].
        u32x4 g0 = { 1u,
                     ldsoff,
                     (uint32_t)gaddr,
                     (uint32_t)((gaddr >> 32) & 0x01FFFFFFu) | 0x80000000u };

        // D# group 1: workgroup_mask=0 (not in a cluster), data_size=2 (4B),
        // tensor_dim0=256, tensor_dim1=1, tile_dim0=256, tile_dim1=1,
        // tile_dim2=0, tensor_dim0_stride=256, tensor_dim1_stride=256.
        i32x8 g1 = { (2 << 16),          // [17:16] data_size = 4 bytes
                     (LUT_N << 16),      // [63:48] tensor_dim0 lo16
                     (1 << 16),          // [95:80] tensor_dim1 = 1
                     (LUT_N << 16),      // [127:112] tile_dim0 = 256
                     1,                  // [143:128] tile_dim1 = 1; tile_dim2 = 0
                     LUT_N,              // [191:160] tensor_dim0_stride lo32
                     (LUT_N << 16),      // [223:208] tensor_dim1_stride lo16
                     0 };
        i32x4 g2 = { 0, 0, 0, 0 };       // tensor_dim2.. unused (<=2D tile)
        i32x4 g3 = { 0, 0, 0, 0 };
        i32x8 g4 = { 0, 0, 0, 0, 0, 0, 0, 0 };  // extra operand (clang-23 6-arg form)

        __builtin_amdgcn_tensor_load_to_lds(g0, g1, g2, g3, g4, /*cpol=*/0);
        __builtin_amdgcn_s_wait_tensorcnt(0);
    }
    __syncthreads();

    // --- Streaming map: 4 x float4 per thread, coalesced b128 accesses.
    const size_t base = (size_t)slice * IMG_HW + (size_t)chunk * ELEMS_PER_BLOCK;
    const v4f* __restrict__ xin = (const v4f*)(x + base);
    v4f* __restrict__ o         = (v4f*)(out + base);

    #pragma unroll
    for (int i = 0; i < V4_PER_THREAD; ++i) {
        const uint32_t k = threadIdx.x + i * TPB;
        v4f v = __builtin_nontemporal_load(&xin[k]);   // streaming read
        // jnp.round == round-half-to-even -> v_rndne_f32 via rintf.
        const int ix = (int)fminf(fmaxf(__builtin_rintf(v.x * 255.0f), 0.0f), 255.0f);
        const int iy = (int)fminf(fmaxf(__builtin_rintf(v.y * 255.0f), 0.0f), 255.0f);
        const int iz = (int)fminf(fmaxf(__builtin_rintf(v.z * 255.0f), 0.0f), 255.0f);
        const int iw = (int)fminf(fmaxf(__builtin_rintf(v.w * 255.0f), 0.0f), 255.0f);
        v4f r;
        r.x = lut[ix];                                  // ds_load_b32 gather
        r.y = lut[iy];
        r.z = lut[iz];
        r.w = lut[iw];
        __builtin_nontemporal_store(r, &o[k]);          // streaming write
    }
}

extern "C" void kernel_launch(void* const* d_in, const int* in_sizes, int n_in,
                              void* d_out, int out_size, void* d_ws, size_t ws_size,
                              hipStream_t stream) {
    (void)in_sizes; (void)n_in; (void)out_size; (void)d_ws; (void)ws_size;
    const float* x     = (const float*)d_in[0];   // [64,3,512,512] f32
    const float* table = (const float*)d_in[1];   // [64,3,256]     f32
    float* out         = (float*)d_out;           // [64,3,512,512] f32

    dim3 grid(CHUNKS_PER_SLICE, N_SLICES);        // 64 x 192 blocks
    RandomPixelMapping_lut_kernel<<<grid, TPB, 0, stream>>>(x, table, out);
}